// Wavefront_SoS_7490422964466
// MI455X (gfx1250) — compile-verified
//
#include <hip/hip_runtime.h>

typedef __attribute__((ext_vector_type(2))) float v2f;
typedef __attribute__((ext_vector_type(8))) float v8f;

#define N_GRID   4096
#define N_POINTS 2048
#define N_INT    4096
#define CHUNKS   32
#define SPC      (N_INT / CHUNKS)   // 128 samples per chunk
#define SPP      (SPC / 4)          // 32 samples per (lane, accumulator) partial

// One wave handles 16 thetas x one 128-sample chunk.
// Each theta's 4 partials sit in the 16x4 fp32 WMMA A-matrix layout:
//   lanes 0-15 : acc0 -> K=0, acc1 -> K=1   (theta = group*16 + lane)
//   lanes 16-31: acc0 -> K=2, acc1 -> K=3   (theta = group*16 + lane-16)
// v_wmma_f32_16x16x4_f32 against all-ones B reduces K for all 16 thetas at once.
__global__ __launch_bounds__(256) void ray_integrate(
    const float* __restrict__ xs, const float* __restrict__ ys,
    const float* __restrict__ SoS, const float* __restrict__ xv,
    const float* __restrict__ yv, float* __restrict__ part)
{
    const int wave  = blockIdx.x * (blockDim.x >> 5) + (threadIdx.x >> 5);
    const int lane  = threadIdx.x & 31;
    const int tg    = wave / CHUNKS;     // theta group: 0..127
    const int chunk = wave % CHUNKS;     // sample chunk: 0..31
    const int m     = lane & 15;
    const int half  = lane >> 4;

    const float x  = xs[0];
    const float y  = ys[0];
    const float x0 = xv[0];
    const float dx = xv[1] - xv[0];
    const float y0 = yv[0];
    const float dy = yv[1] - yv[0];

    const float Rb      = 0.04f;
    const float v0c     = 1500.0f;
    const float twopi   = 6.28318530717958647692f;
    const float inv4095 = 1.0f / 4095.0f;            // 1/(N_INT-1), also linspace step
    const float invdx   = 1.0f / dx;
    const float invdy   = 1.0f / dy;

    // Per-lane ray geometry (theta for this lane's row of the A-matrix)
    const int   t     = tg * 16 + m;
    const float theta = (twopi / 2047.0f) * (float)t;

    const float r    = sqrtf(x * x + y * y);
    const float phi  = atan2f(x, y);                  // note arg order (x, y) as in reference
    const float sd   = r * sinf(theta - phi);
    const float disc = fmaxf(Rb * Rb - sd * sd, 0.0f);
    const float sq   = sqrtf(disc);
    const float ct   = cosf(theta - phi);
    const float l_in  = sq + r * ct;
    const float l_out = 2.0f * sq * (ct >= 0.0f ? 1.0f : 0.0f);
    const float l     = (r < Rb) ? l_in : l_out;
    const float sinT  = sinf(theta);
    const float cosT  = cosf(theta);
    const float h     = l * inv4095;                  // constant trapezoid spacing per ray

    auto sampleF = [&](int s) -> float {
        float stepf = (float)s * inv4095;
        float ls    = l * stepf;
        float fx    = (x - ls * sinT - x0) * invdx;
        float fy    = (y - ls * cosT - y0) * invdy;
        int   xi    = (int)rintf(fx);                 // v_rndne: round half-to-even
        int   yi    = (int)rintf(fy);
        xi = min(max(xi, 0), N_GRID - 1);             // JAX clamps OOB gathers
        int row = (-yi) & (N_GRID - 1);               // (-yi) mod 4096
        float val = SoS[(row << 12) + xi];
        // f = 1 - v0/val via v_rcp_f32 + one Newton step (rel err ~1e-9)
        float rc = __builtin_amdgcn_rcpf(val);
        rc = fmaf(fmaf(-val, rc, 1.0f), rc, rc);
        return fmaf(-v0c, rc, 1.0f);
    };

    const int k0 = half * 2;
    const int s0 = chunk * SPC + k0 * SPP;            // acc0 range start
    const int s1 = s0 + SPP;                          // acc1 range start

    float acc0 = 0.0f, acc1 = 0.0f;
#pragma unroll 4
    for (int i = 0; i < SPP; ++i) {
        acc0 += sampleF(s0 + i);
        acc1 += sampleF(s1 + i);
    }

    // Trapezoid endpoint corrections: -0.5*f(0) lives in partial (chunk0,K0),
    // -0.5*f(N-1) in partial (chunk31,K3). Divergent here; reconverges before WMMA.
    if (chunk == 0 && half == 0)          acc0 -= 0.5f * sampleF(0);
    if (chunk == CHUNKS - 1 && half == 1) acc1 -= 0.5f * sampleF(N_INT - 1);

    // Fold per-ray spacing so the WMMA output is the final partial wf.
    acc0 *= h;
    acc1 *= h;

    // Cross-lane reduction: D[m][*] = acc_K0 + acc_K1 + acc_K2 + acc_K3 for theta m.
    v2f a; a.x = acc0; a.y = acc1;
    v2f b; b.x = 1.0f; b.y = 1.0f;
    v8f c = {};
    v8f d = __builtin_amdgcn_wmma_f32_16x16x4_f32(
        /*neg_a=*/false, a, /*neg_b=*/false, b,
        /*c_mod=*/(short)0, c, /*reuse_a=*/false, /*reuse_b=*/false);

    // D layout: VGPR v holds row v (lanes 0-15) / row v+8 (lanes 16-31), all cols equal.
    // Flat select chain (v_cndmask x7) + one predicated store: lane v -> theta v,
    // lane 16+v -> theta v+8.
    float w = d[0];
    w = (m == 1) ? d[1] : w;
    w = (m == 2) ? d[2] : w;
    w = (m == 3) ? d[3] : w;
    w = (m == 4) ? d[4] : w;
    w = (m == 5) ? d[5] : w;
    w = (m == 6) ? d[6] : w;
    w = (m == 7) ? d[7] : w;
    if (m < 8) {
        part[(size_t)(tg * 16 + m + 8 * half) * CHUNKS + chunk] = w;
    }
}

__global__ __launch_bounds__(256) void reduce_out(
    const float* __restrict__ part, float* __restrict__ out)
{
    int t = blockIdx.x * blockDim.x + threadIdx.x;
    if (t >= N_POINTS) return;
    float s = 0.0f;
#pragma unroll
    for (int c = 0; c < CHUNKS; ++c) s += part[(size_t)t * CHUNKS + c];
    out[t]            = (6.28318530717958647692f / 2047.0f) * (float)t;  // thetas
    out[N_POINTS + t] = s;                                               // wf
}

extern "C" void kernel_launch(void* const* d_in, const int* in_sizes, int n_in,
                              void* d_out, int out_size, void* d_ws, size_t ws_size,
                              hipStream_t stream) {
    const float* xs  = (const float*)d_in[0];
    const float* ys  = (const float*)d_in[1];
    const float* SoS = (const float*)d_in[2];
    const float* xv  = (const float*)d_in[3];
    const float* yv  = (const float*)d_in[4];
    float* out  = (float*)d_out;
    float* part = (float*)d_ws;   // N_POINTS * CHUNKS floats = 256 KB

    // 128 theta-groups x 32 chunks = 4096 waves; 256 threads (8 waves) per block.
    hipLaunchKernelGGL(ray_integrate, dim3((N_POINTS / 16) * CHUNKS / 8), dim3(256),
                       0, stream, xs, ys, SoS, xv, yv, part);
    hipLaunchKernelGGL(reduce_out, dim3(N_POINTS / 256), dim3(256),
                       0, stream, part, out);
}